// TrustRegionBase_29480655519962
// MI455X (gfx1250) — compile-verified
//
#include <hip/hip_runtime.h>
#include <math.h>

typedef float v2f __attribute__((ext_vector_type(2)));
typedef float v8f __attribute__((ext_vector_type(8)));

// Problem constants (from reference header)
constexpr int CH = 16;
constexpr int Hh = 240;
constexpr int Ww = 320;
constexpr int HW  = Hh * Ww;
constexpr int CHW = CH * HW;
constexpr int NCHUNK = CHW / 32;         // 32 samples per wave-chunk
constexpr int OSTRIDE = 12 + CHW;        // per-batch output row
constexpr int WAVES = 8;                 // waves per block (wave32)
constexpr int TPB = WAVES * 32;          // 256 threads
constexpr int GX = 300;                  // blocks per batch -> 2400 waves/batch

#define OCC_THRES 0.1f
#define HUBER     0.2f
#define LM_DAMP   1e-6f

__global__ void init_kernel(const float* __restrict__ Rin, const float* __restrict__ tin,
                            float* __restrict__ pose, float* __restrict__ Hacc,
                            float* __restrict__ gacc, int B)
{
    int i = threadIdx.x;
    for (int k = i; k < B * 12; k += blockDim.x) {
        int b = k / 12, j = k % 12;
        pose[k] = (j < 9) ? Rin[b * 9 + j] : tin[b * 3 + (j - 9)];
    }
    for (int k = i; k < B * 36; k += blockDim.x) Hacc[k] = 0.f;
    for (int k = i; k < B * 6;  k += blockDim.x) gacc[k] = 0.f;
}

// One streaming pass: per element compute Sobel grad, warp Jacobian J, warped
// residual r, weight w.  Each wave stages 32 samples into LDS, then 8x
// V_WMMA_F32_16X16X4_F32 accumulate the 6x7 block [JtWJ | JtWr] (A rows = J
// components, B cols 0..5 = w*J, col 6 = w*r, K = sample index).
template<bool DO_H, bool FIRST>
__global__ __launch_bounds__(TPB)
void irls_pass(const float* __restrict__ x0, const float* __restrict__ x1,
               const float* __restrict__ invD0, const float* __restrict__ invD1,
               const float* __restrict__ Kin, const float* __restrict__ wPrior,
               const float* __restrict__ f0, const float* __restrict__ pose,
               float* __restrict__ Hacc, float* __restrict__ gacc,
               float* __restrict__ out)
{
    __shared__ float sA[WAVES][32][17];  // stride 17: conflict-free writes & reads
    __shared__ float sB[WAVES][32][17];

    const int b = blockIdx.y;
    const float fx = Kin[b * 4 + 0], fy = Kin[b * 4 + 1];
    const float cx = Kin[b * 4 + 2], cy = Kin[b * 4 + 3];
    const float* P = pose + b * 12;
    const float R00 = P[0], R01 = P[1], R02 = P[2];
    const float R10 = P[3], R11 = P[4], R12 = P[5];
    const float R20 = P[6], R21 = P[7], R22 = P[8];
    const float t0 = P[9], t1 = P[10], t2 = P[11];

    const int lane = threadIdx.x & 31;
    const int wv   = threadIdx.x >> 5;
    const int gw   = blockIdx.x * WAVES + wv;
    const int nw   = gridDim.x * WAVES;
    const int half = lane >> 4;     // lane half selects K in {0,1}/{2,3} (A) and rows K/K+2 (B)
    const int m16  = lane & 15;     // A row M / B col N

    const float* x0b = x0 + (size_t)b * CHW;
    const float* x1b = x1 + (size_t)b * CHW;
    const float* d0b = invD0 + (size_t)b * HW;
    const float* d1b = invD1 + (size_t)b * HW;
    const float* f0b = f0 + (size_t)b * CHW;
    float* outb = out + (size_t)b * OSTRIDE + 12;   // weights region

    // zero-fill the padded tails once (never overwritten)
    float* aRow = sA[wv][lane];
    float* bRow = sB[wv][lane];
    #pragma unroll
    for (int q = 6; q < 16; q++) aRow[q] = 0.f;
    #pragma unroll
    for (int q = 7; q < 16; q++) bRow[q] = 0.f;

    v8f acc = {0.f, 0.f, 0.f, 0.f, 0.f, 0.f, 0.f, 0.f};

    for (int chunk = gw; chunk < NCHUNK; chunk += nw) {
        const int s = chunk * 32 + lane;
        const int c = s / HW;
        const int p = s - c * HW;
        const int iy = p / Ww;
        const int ix = p - iy * Ww;
        const float* img = x0b + c * HW;

        if (chunk + nw < NCHUNK)   // prefetch next chunk of x0 (global_prefetch_b8)
            __builtin_prefetch(x0b + (s + nw * 32), 0, 0);

        // --- Sobel gradient of x0, edge-clamped, normalized ---
        int xm = ix > 0 ? ix - 1 : 0,      xp = ix < Ww - 1 ? ix + 1 : Ww - 1;
        int ym = iy > 0 ? iy - 1 : 0,      yp = iy < Hh - 1 ? iy + 1 : Hh - 1;
        float i00 = img[ym * Ww + xm], i01 = img[ym * Ww + ix], i02 = img[ym * Ww + xp];
        float i10 = img[iy * Ww + xm], ic  = img[iy * Ww + ix], i12 = img[iy * Ww + xp];
        float i20 = img[yp * Ww + xm], i21 = img[yp * Ww + ix], i22 = img[yp * Ww + xp];
        float gx = (i02 - i00) + 2.f * (i12 - i10) + (i22 - i20);
        float gy = (i20 - i00) + 2.f * (i21 - i01) + (i22 - i02);
        float mag = sqrtf(gx * gx + gy * gy + 1e-8f);
        gx /= mag; gy /= mag;

        // --- warp Jacobian (J = gx*fx*dx_dp + gy*fy*dy_dp) ---
        float xn = ((float)ix - cx) / fx;
        float yn = ((float)iy - cy) / fy;
        float invd = d0b[p];
        float xy = xn * yn;
        float J0 = gx * (-xy) * fx          + gy * (-1.f - yn * yn) * fy;
        float J1 = gx * (1.f + xn * xn) * fx + gy * (xy) * fy;
        float J2 = gx * (-yn) * fx          + gy * (xn) * fy;
        float J3 = gx * invd * fx;
        float J4 = gy * invd * fy;
        float J5 = gx * (-invd * xn) * fx   + gy * (-invd * yn) * fy;

        // --- warped residual ---
        float X = R00 * xn + R01 * yn + R02 + t0 * invd;
        float Y = R10 * xn + R11 * yn + R12 + t1 * invd;
        float S = R20 * xn + R21 * yn + R22 + t2 * invd;
        float invS = 1.f / S;
        float u = X * invS * fx + cx;
        float v = Y * invS * fy + cy;
        float inv_z = invd * invS;

        float uc = fminf(fmaxf(u, 0.f), (float)(Ww - 1));
        float vc = fminf(fmaxf(v, 0.f), (float)(Hh - 1));
        float u0f = floorf(uc), v0f = floorf(vc);
        float wu = uc - u0f, wvv = vc - v0f;
        int u0 = (int)u0f, v0 = (int)v0f;
        int u1 = u0 + 1 < Ww ? u0 + 1 : Ww - 1;
        int v1 = v0 + 1 < Hh ? v0 + 1 : Hh - 1;

        const float* F1 = x1b + c * HW;
        float fa = F1[v0 * Ww + u0], fb = F1[v0 * Ww + u1];
        float fc = F1[v1 * Ww + u0], fd = F1[v1 * Ww + u1];
        float x1w = (1.f - wvv) * ((1.f - wu) * fa + wu * fb)
                  +        wvv  * ((1.f - wu) * fc + wu * fd);
        float da = d1b[v0 * Ww + u0], db = d1b[v0 * Ww + u1];
        float dc = d1b[v1 * Ww + u0], dd = d1b[v1 * Ww + u1];
        float d1w = (1.f - wvv) * ((1.f - wu) * da + wu * db)
                  +        wvv  * ((1.f - wu) * dc + wu * dd);

        bool inview = (inv_z > d1w - OCC_THRES) &&
                      (u > 0.f) && (u < (float)(Ww - 1)) &&
                      (v > 0.f) && (v < (float)(Hh - 1));
        float r = inview ? (x1w - ic) : 0.001f;
        r *= (1.f + f0b[c * HW + p]);

        float w;
        if (FIRST) {
            w = wPrior[(size_t)b * CHW + s] * fminf(1.f, HUBER / (fabsf(r) + 1e-8f));
            outb[s] = w;             // weights are the output & reused by later passes
        } else {
            w = outb[s];
        }

        // --- stage into LDS (per-wave region) ---
        aRow[0] = J0; aRow[1] = J1; aRow[2] = J2;
        aRow[3] = J3; aRow[4] = J4; aRow[5] = J5;
        bRow[0] = w * J0; bRow[1] = w * J1; bRow[2] = w * J2;
        bRow[3] = w * J3; bRow[4] = w * J4; bRow[5] = w * J5;
        bRow[6] = w * r;
        __builtin_amdgcn_wave_barrier();

        // --- 8 WMMAs consume the 32 staged samples (K=4 each) ---
        #pragma unroll
        for (int j = 0; j < 8; j++) {
            int sa = 4 * j + (half << 1);      // sample for K=0 (half 0) / K=2 (half 1)
            v2f a, bb;
            a[0]  = sA[wv][sa    ][m16];       // A[M=m16, K]   (f32 16x4 layout)
            a[1]  = sA[wv][sa + 1][m16];       // A[M=m16, K+1]
            bb[0] = sB[wv][sa    ][m16];       // B[K,   N=m16] (f32 4x16 layout)
            bb[1] = sB[wv][sa + 1][m16];       // B[K+1, N=m16]
            acc = __builtin_amdgcn_wmma_f32_16x16x4_f32(
                      false, a, false, bb, (short)0, acc, false, false);
        }
        __builtin_amdgcn_wave_barrier();
    }

    // D element at (M = reg + 8*half, N = m16); only rows 0..5, cols 0..6 are live.
    if (half == 0 && m16 < 7) {
        #pragma unroll
        for (int reg = 0; reg < 6; reg++) {
            float val = acc[reg];
            if (m16 < 6) {
                if (DO_H) atomicAdd(&Hacc[b * 36 + reg * 6 + m16], val);  // JtWJ
            } else {
                atomicAdd(&gacc[b * 6 + reg], val);                       // JtWr
            }
        }
    }
}

// Per-batch: (JtWJ + damp*I) xi = JtWr  ->  exp_se3(xi)  ->  pose update.
__global__ void solve_kernel(const float* __restrict__ Hacc, float* __restrict__ gacc,
                             float* __restrict__ pose, float* __restrict__ out,
                             int B, int writeOut)
{
    int b = blockIdx.x * blockDim.x + threadIdx.x;
    if (b >= B) return;

    float M[6][7];
    for (int i = 0; i < 6; i++) {
        for (int j = 0; j < 6; j++)
            M[i][j] = Hacc[b * 36 + i * 6 + j] + (i == j ? LM_DAMP : 0.f);
        M[i][6] = gacc[b * 6 + i];
    }
    // Gaussian elimination with partial pivoting
    for (int k = 0; k < 6; k++) {
        int piv = k; float best = fabsf(M[k][k]);
        for (int r2 = k + 1; r2 < 6; r2++) {
            float a2 = fabsf(M[r2][k]);
            if (a2 > best) { best = a2; piv = r2; }
        }
        if (piv != k)
            for (int j = 0; j < 7; j++) { float tp = M[k][j]; M[k][j] = M[piv][j]; M[piv][j] = tp; }
        float inv = 1.f / M[k][k];
        for (int r2 = k + 1; r2 < 6; r2++) {
            float f = M[r2][k] * inv;
            for (int j = k; j < 7; j++) M[r2][j] -= f * M[k][j];
        }
    }
    float xi[6];
    for (int k = 5; k >= 0; k--) {
        float s2 = M[k][6];
        for (int j = k + 1; j < 6; j++) s2 -= M[k][j] * xi[j];
        xi[k] = s2 / M[k][k];
    }

    // exp_se3
    float wx = xi[0], wy = xi[1], wz = xi[2], vx = xi[3], vy = xi[4], vz = xi[5];
    float th2 = wx * wx + wy * wy + wz * wz;
    float th = sqrtf(th2 + 1e-12f);
    float Ac, Bc, Cc;
    if (th < 1e-4f) {
        Ac = 1.f - th2 / 6.f; Bc = 0.5f - th2 / 24.f; Cc = 1.f / 6.f - th2 / 120.f;
    } else {
        float sn = sinf(th), cs = cosf(th);
        Ac = sn / th;
        Bc = (1.f - cs) / (th2 + 1e-12f);
        Cc = (th - sn) / (th2 * th + 1e-12f);
    }
    float Wm[3][3] = {{0.f, -wz, wy}, {wz, 0.f, -wx}, {-wy, wx, 0.f}};
    float W2[3][3];
    for (int i = 0; i < 3; i++)
        for (int j = 0; j < 3; j++) {
            float s3 = 0.f;
            for (int k = 0; k < 3; k++) s3 += Wm[i][k] * Wm[k][j];
            W2[i][j] = s3;
        }
    float dR[3][3], V[3][3];
    for (int i = 0; i < 3; i++)
        for (int j = 0; j < 3; j++) {
            float id = (i == j) ? 1.f : 0.f;
            dR[i][j] = id + Ac * Wm[i][j] + Bc * W2[i][j];
            V[i][j]  = id + Bc * Wm[i][j] + Cc * W2[i][j];
        }
    float dt[3];
    for (int i = 0; i < 3; i++)
        dt[i] = V[i][0] * vx + V[i][1] * vy + V[i][2] * vz;

    float* P = pose + b * 12;
    float R[3][3] = {{P[0], P[1], P[2]}, {P[3], P[4], P[5]}, {P[6], P[7], P[8]}};
    float t[3] = {P[9], P[10], P[11]};
    float Rn[3][3];
    for (int i = 0; i < 3; i++)
        for (int j = 0; j < 3; j++) {      // Rn = R * dR^T
            float s3 = 0.f;
            for (int k = 0; k < 3; k++) s3 += R[i][k] * dR[j][k];
            Rn[i][j] = s3;
        }
    float tn[3];
    for (int i = 0; i < 3; i++) {          // tn = t - Rn * dt
        float s3 = 0.f;
        for (int j = 0; j < 3; j++) s3 += Rn[i][j] * dt[j];
        tn[i] = t[i] - s3;
    }
    for (int i = 0; i < 3; i++)
        for (int j = 0; j < 3; j++) P[i * 3 + j] = Rn[i][j];
    P[9] = tn[0]; P[10] = tn[1]; P[11] = tn[2];

    for (int i = 0; i < 6; i++) gacc[b * 6 + i] = 0.f;   // re-arm JtWr for next pass

    if (writeOut) {
        float* o = out + (size_t)b * OSTRIDE;
        for (int i = 0; i < 12; i++) o[i] = P[i];
    }
}

extern "C" void kernel_launch(void* const* d_in, const int* in_sizes, int n_in,
                              void* d_out, int out_size, void* d_ws, size_t ws_size,
                              hipStream_t stream)
{
    (void)n_in; (void)out_size; (void)ws_size;
    const float* Rin    = (const float*)d_in[0];
    const float* tin    = (const float*)d_in[1];
    const float* x0     = (const float*)d_in[2];
    const float* x1     = (const float*)d_in[3];
    const float* iD0    = (const float*)d_in[4];
    const float* iD1    = (const float*)d_in[5];
    const float* Kin    = (const float*)d_in[6];
    const float* wPrior = (const float*)d_in[7];
    const float* f0     = (const float*)d_in[8];
    float* out = (float*)d_out;

    const int B = in_sizes[1] / 3;

    float* pose = (float*)d_ws;           // B*12
    float* Hacc = pose + B * 12;          // B*36
    float* gacc = Hacc + B * 36;          // B*6

    init_kernel<<<1, 256, 0, stream>>>(Rin, tin, pose, Hacc, gacc, B);

    dim3 grid(GX, B);
    // pass 1: weights + JtWJ + JtWr(initial pose)
    irls_pass<true, true><<<grid, TPB, 0, stream>>>(
        x0, x1, iD0, iD1, Kin, wPrior, f0, pose, Hacc, gacc, out);
    solve_kernel<<<1, 32, 0, stream>>>(Hacc, gacc, pose, out, B, 0);
    // pass 2: JtWr with updated pose
    irls_pass<false, false><<<grid, TPB, 0, stream>>>(
        x0, x1, iD0, iD1, Kin, wPrior, f0, pose, Hacc, gacc, out);
    solve_kernel<<<1, 32, 0, stream>>>(Hacc, gacc, pose, out, B, 0);
    // pass 3: JtWr with updated pose; final solve writes R|t to output
    irls_pass<false, false><<<grid, TPB, 0, stream>>>(
        x0, x1, iD0, iD1, Kin, wPrior, f0, pose, Hacc, gacc, out);
    solve_kernel<<<1, 32, 0, stream>>>(Hacc, gacc, pose, out, B, 1);
}